// MakeUpLoss_11931419148728
// MI455X (gfx1250) — compile-verified
//
#include <hip/hip_runtime.h>
#include <hip/hip_bf16.h>

typedef unsigned int uint;
typedef uint __attribute__((ext_vector_type(4))) v4u;
typedef int  __attribute__((ext_vector_type(4))) v4i;
typedef int  __attribute__((ext_vector_type(8))) v8i;

#define BB   4
#define CC   3
#define HH   512
#define WW   512
#define HW   (HH * WW)          // 262144
#define KBIN 33
#define NREG 3
#define NHIST (NREG * CC * KBIN) // 297
#define CHUNK 1024
#define NCHUNK (HW / CHUNK)     // 256
#define NWGX  32
#define NWAVE 8                 // 256 threads / wave32

#ifndef HAVE_TDM
#if defined(__has_builtin)
#if __has_builtin(__builtin_amdgcn_tensor_load_to_lds) && __has_builtin(__builtin_amdgcn_s_wait_tensorcnt)
#define HAVE_TDM 1
#endif
#endif
#endif
#ifndef HAVE_TDM
#define HAVE_TDM 0
#endif

#if HAVE_TDM
// Issue a 1-D TDM copy of `words` dwords from global `gp` into LDS byte
// address `lds_addr`.  D# layout per CDNA5 ISA ch.8:
//  group0: [1:0]=count=1, [63:32]=lds_addr, [120:64]=global_addr, [127:126]=type=2
//  group1: [17:16]=data_size(2=4B), [79:48]=tensor_dim0, [111:80]=tensor_dim1=1,
//          [127:112]=tile_dim0, [143:128]=tile_dim1=1, [207:160]=dim0_stride
__device__ __forceinline__ void tdm_load_1d(uint lds_addr, const void* gp, int words) {
    unsigned long long ga = (unsigned long long)(size_t)gp;
    v4u g0;
    g0.x = 1u;                                   // count = 1 (valid descriptor)
    g0.y = lds_addr;                             // LDS byte address
    g0.z = (uint)(ga & 0xFFFFFFFFu);             // global_addr[31:0]
    g0.w = (uint)((ga >> 32) & 0x1FFFFFFu) | 0x80000000u; // addr[56:32] | type=2
    v8i g1;
    g1[0] = 0x00020000;                          // workgroup_mask=0, data_size=4B
    g1[1] = (words & 0xFFFF) << 16;              // tensor_dim0[15:0]
    g1[2] = ((words >> 16) & 0xFFFF) | (1 << 16);// tensor_dim0[31:16] | tensor_dim1=1
    g1[3] = (words & 0xFFFF) << 16;              // tile_dim0 = words
    g1[4] = 1;                                   // tile_dim1 = 1, tile_dim2 = 0
    g1[5] = words;                               // tensor_dim0_stride[31:0]
    g1[6] = 0;
    g1[7] = 0;
    v4i g2 = (v4i)(0);
    v4i g3 = (v4i)(0);
#if __clang_major__ >= 23
    v8i g4 = (v8i)(0);
    __builtin_amdgcn_tensor_load_to_lds(g0, g1, g2, g3, g4, 0);
#else
    __builtin_amdgcn_tensor_load_to_lds(g0, g1, g2, g3, 0);
#endif
}
#endif

// ---------------------------------------------------------------------------
// Kernel 1: zero the workspace accumulators (N counts + global histograms)
// ---------------------------------------------------------------------------
__global__ void mk_init_kernel(float* g_N, float* g_hist) {
    int i = blockIdx.x * 256 + threadIdx.x;
    if (i < 2 * BB * NREG) g_N[i] = 0.0f;
    if (i < 2 * BB * NHIST) g_hist[i] = 0.0f;
}

// ---------------------------------------------------------------------------
// Kernel 2: eye bounding box per (image-side z, batch b), with the reference's
// 15% pad + double-pad-on-max quirk. One block per (z,b).
// ---------------------------------------------------------------------------
__global__ __launch_bounds__(256) void mk_bbox_kernel(const int* __restrict__ seg_gen,
                                                      const int* __restrict__ seg_real,
                                                      int* __restrict__ bbox) {
    const int zb = blockIdx.x;            // 0..7
    const int z = zb >> 2, b = zb & 3;
    const int* seg = (z ? seg_real : seg_gen) + b * HW;
    const int tid = threadIdx.x;

    int ymin = HH, ymax = -1, xmin = WW, xmax = -1;
    for (int p = tid; p < HW; p += 256) {
        int s = seg[p];
        if (s == 4 || s == 5) {
            int y = p >> 9, x = p & (WW - 1);
            ymin = min(ymin, y); ymax = max(ymax, y);
            xmin = min(xmin, x); xmax = max(xmax, x);
        }
    }
    __shared__ int sm0[256], sm1[256], sm2[256], sm3[256];
    sm0[tid] = ymin; sm1[tid] = ymax; sm2[tid] = xmin; sm3[tid] = xmax;
    __syncthreads();
    for (int s = 128; s > 0; s >>= 1) {
        if (tid < s) {
            sm0[tid] = min(sm0[tid], sm0[tid + s]);
            sm1[tid] = max(sm1[tid], sm1[tid + s]);
            sm2[tid] = min(sm2[tid], sm2[tid + s]);
            sm3[tid] = max(sm3[tid], sm3[tid + s]);
        }
        __syncthreads();
    }
    if (tid == 0) {
        int y0 = sm0[0], y1 = sm1[0], x0 = sm2[0], x1 = sm3[0];
        bool has = (y1 >= 0);
        int py = (int)(0.15f * (float)(y1 - y0));   // trunc toward zero (== astype)
        int px = (int)(0.15f * (float)(x1 - x0));
        int ylo = max(0, y0 - py);
        int yhi = min(HH, y1 + py);
        yhi = min(HH - 1, yhi + py);                // double-pad quirk
        int xlo = max(0, x0 - px);
        int xhi = min(WW, x1 + px);
        xhi = min(WW - 1, xhi + px);
        if (!has) { ylo = 0; yhi = 0; xlo = 0; xhi = 0; }
        int* o = bbox + zb * 4;
        o[0] = ylo; o[1] = yhi; o[2] = xlo; o[3] = xhi;
    }
}

// ---------------------------------------------------------------------------
// Kernel 3: fused single-pass masked soft-histogram.  grid = (NWGX, 8).
// Each pixel contributes linear weights (1-t, t) to its two adjacent bins
// (the SPACING factor cancels in num/(N*S)).  Chunks of 1024 pixels
// (seg + 3 channels) are double-buffered into LDS by the Tensor Data Mover;
// histograms are privatized per wave to cut ds_add_f32 contention 8x.
// ---------------------------------------------------------------------------
__global__ __launch_bounds__(256) void mk_hist_kernel(const float* __restrict__ gen,
                                                      const float* __restrict__ realm,
                                                      const int* __restrict__ seg_gen,
                                                      const int* __restrict__ seg_real,
                                                      const int* __restrict__ bbox,
                                                      float* __restrict__ g_hist,
                                                      float* __restrict__ g_N) {
    __shared__ uint  stage[2][4][CHUNK];   // [buf][seg,c0,c1,c2][pixel]  (32 KB)
    __shared__ float hl[NWAVE][NHIST];     // per-wave [region][channel][bin] (9.5 KB)
    __shared__ float nl[NREG];

    const int tid  = threadIdx.x;
    const int wave = tid >> 5;
    const int zb   = blockIdx.y;           // 0..7
    const int z = zb >> 2, b = zb & 3;
    const uint*  segp = (const uint*)((z ? seg_real : seg_gen) + b * HW);
    const float* imgp = (z ? realm : gen) + (size_t)b * CC * HW;

    for (int i = tid; i < NWAVE * NHIST; i += 256) hl[0][i] = 0.0f;
    if (tid < NREG) nl[tid] = 0.0f;
    const int by0 = bbox[zb * 4 + 0], by1 = bbox[zb * 4 + 1];
    const int bx0 = bbox[zb * 4 + 2], bx1 = bbox[zb * 4 + 3];
    __syncthreads();

    float lN0 = 0.0f, lN1 = 0.0f, lN2 = 0.0f;
    const int c0 = blockIdx.x;
    int bi = 0;

#if HAVE_TDM
    if (tid < 32) {   // wave 0 drives the TDM pipeline (wave-uniform guard)
        const uint off0 = (uint)c0 * CHUNK;
        tdm_load_1d((uint)(size_t)&stage[0][0][0], segp + off0, CHUNK);
        tdm_load_1d((uint)(size_t)&stage[0][1][0], imgp + off0, CHUNK);
        tdm_load_1d((uint)(size_t)&stage[0][2][0], imgp + HW + off0, CHUNK);
        tdm_load_1d((uint)(size_t)&stage[0][3][0], imgp + 2 * HW + off0, CHUNK);
    }
#endif

    for (int ch = c0; ch < NCHUNK; ch += NWGX) {
#if HAVE_TDM
        const int nxt = ch + NWGX;
        if (tid < 32) {
            if (nxt < NCHUNK) {
                const uint off = (uint)nxt * CHUNK;
                const int nb = bi ^ 1;
                tdm_load_1d((uint)(size_t)&stage[nb][0][0], segp + off, CHUNK);
                tdm_load_1d((uint)(size_t)&stage[nb][1][0], imgp + off, CHUNK);
                tdm_load_1d((uint)(size_t)&stage[nb][2][0], imgp + HW + off, CHUNK);
                tdm_load_1d((uint)(size_t)&stage[nb][3][0], imgp + 2 * HW + off, CHUNK);
                __builtin_amdgcn_s_wait_tensorcnt(4);  // current chunk's 4 DMAs done
            } else {
                __builtin_amdgcn_s_wait_tensorcnt(0);
            }
        }
        __syncthreads();
#else
        {   // fallback: cooperative copy into the active buffer
            const uint off = (uint)ch * CHUNK;
            const uint* s0 = segp + off;
            const uint* s1 = (const uint*)(imgp + off);
            const uint* s2 = (const uint*)(imgp + HW + off);
            const uint* s3 = (const uint*)(imgp + 2 * HW + off);
            for (int i = tid; i < CHUNK; i += 256) {
                stage[bi][0][i] = s0[i];
                stage[bi][1][i] = s1[i];
                stage[bi][2][i] = s2[i];
                stage[bi][3][i] = s3[i];
            }
            __syncthreads();
        }
#endif
        // Each thread owns 4 consecutive staged pixels -> one ds_load_b128 per array.
        const int p0 = ch * CHUNK + tid * 4;
        const uint4 sv4 = *reinterpret_cast<const uint4*>(&stage[bi][0][tid * 4]);
        const uint4 va4 = *reinterpret_cast<const uint4*>(&stage[bi][1][tid * 4]);
        const uint4 vb4 = *reinterpret_cast<const uint4*>(&stage[bi][2][tid * 4]);
        const uint4 vc4 = *reinterpret_cast<const uint4*>(&stage[bi][3][tid * 4]);
        const uint sva[4] = {sv4.x, sv4.y, sv4.z, sv4.w};
        const uint vaa[4] = {va4.x, va4.y, va4.z, va4.w};
        const uint vba[4] = {vb4.x, vb4.y, vb4.z, vb4.w};
        const uint vca[4] = {vc4.x, vc4.y, vc4.z, vc4.w};
#pragma unroll
        for (int q = 0; q < 4; ++q) {
            const uint sv = sva[q];
            const int p = p0 + q;
            const int y = p >> 9, x = p & (WW - 1);
            const bool mL = (sv == 11u) || (sv == 12u);
            const bool mF = (sv == 1u);
            const bool mE = (y >= by0) & (y < by1) & (x >= bx0) & (x < bx1);
            lN0 += mL ? 1.0f : 0.0f;
            lN1 += mF ? 1.0f : 0.0f;
            lN2 += mE ? 1.0f : 0.0f;
            if (mL | mF | mE) {
                const uint pix[3] = {vaa[q], vba[q], vca[q]};
#pragma unroll
                for (int c = 0; c < CC; ++c) {
                    const float v = __uint_as_float(pix[c]);
                    const float u = (v + 1.0f) * 16.0f;   // 1/SPACING = 16
                    int j = (int)u;
                    j = j < 0 ? 0 : (j > KBIN - 2 ? KBIN - 2 : j);
                    const float t = u - (float)j;
                    const float w0 = 1.0f - t;
                    float* hw = hl[wave];
                    if (mL) { atomicAdd(&hw[(0 * CC + c) * KBIN + j], w0);
                              atomicAdd(&hw[(0 * CC + c) * KBIN + j + 1], t); }
                    if (mF) { atomicAdd(&hw[(1 * CC + c) * KBIN + j], w0);
                              atomicAdd(&hw[(1 * CC + c) * KBIN + j + 1], t); }
                    if (mE) { atomicAdd(&hw[(2 * CC + c) * KBIN + j], w0);
                              atomicAdd(&hw[(2 * CC + c) * KBIN + j + 1], t); }
                }
            }
        }
        __syncthreads();
        bi ^= 1;
    }

    atomicAdd(&nl[0], lN0);
    atomicAdd(&nl[1], lN1);
    atomicAdd(&nl[2], lN2);
    __syncthreads();

    // Collapse the per-wave private histograms and flush once to global.
    float* gh = g_hist + (size_t)zb * NHIST;
    for (int i = tid; i < NHIST; i += 256) {
        float s = 0.0f;
#pragma unroll
        for (int w = 0; w < NWAVE; ++w) s += hl[w][i];
        if (s != 0.0f) atomicAdd(&gh[i], s);
    }
    if (tid < NREG) atomicAdd(&g_N[zb * NREG + tid], nl[tid]);
}

// ---------------------------------------------------------------------------
// Kernel 4: normalize histograms, L1, weights, validity, reduce to scalar.
// ---------------------------------------------------------------------------
__global__ __launch_bounds__(64) void mk_final_kernel(const float* __restrict__ g_hist,
                                                      const float* __restrict__ g_N,
                                                      float* __restrict__ out) {
    __shared__ float red[64];
    const int t = threadIdx.x;
    float contrib = 0.0f;
    if (t < NREG * BB * CC) {                     // 36 items
        const int r = t / (BB * CC);
        const int b = (t / CC) % BB;
        const int c = t % CC;
        const float w = (r == 1) ? 0.1f : 1.0f;   // (lips, face, eye) = (1, 0.1, 1)
        const float* hg = g_hist + ((size_t)(0 * BB + b) * NREG * CC + r * CC + c) * KBIN;
        const float* hr = g_hist + ((size_t)(1 * BB + b) * NREG * CC + r * CC + c) * KBIN;
        const float Ng = g_N[(0 * BB + b) * NREG + r];
        const float Nr = g_N[(1 * BB + b) * NREG + r];
        float Sg = 0.0f, Sr = 0.0f;
        for (int k = 0; k < KBIN; ++k) { Sg += hg[k]; Sr += hr[k]; }
        float dg = Ng * Sg; dg = (dg > 0.0f) ? dg : 1.0f;
        float dr = Nr * Sr; dr = (dr > 0.0f) ? dr : 1.0f;
        float l1 = 0.0f;
        for (int k = 0; k < KBIN; ++k) l1 += fabsf(hg[k] / dg - hr[k] / dr);
        l1 *= (1.0f / (float)KBIN);
        if (Ng > 0.0f && Nr > 0.0f) contrib = w * l1;
    }
    red[t] = contrib;
    __syncthreads();
    for (int s = 32; s > 0; s >>= 1) {
        if (t < s) red[t] += red[t + s];
        __syncthreads();
    }
    if (t == 0) out[0] = red[0] * (1.0f / (float)(BB * CC * 3));
}

// ---------------------------------------------------------------------------
extern "C" void kernel_launch(void* const* d_in, const int* in_sizes, int n_in,
                              void* d_out, int out_size, void* d_ws, size_t ws_size,
                              hipStream_t stream) {
    const float* gen      = (const float*)d_in[0];
    const float* realm    = (const float*)d_in[1];
    const int*   seg_gen  = (const int*)d_in[2];
    const int*   seg_real = (const int*)d_in[3];
    float* out = (float*)d_out;

    char* ws = (char*)d_ws;
    int*   bbox   = (int*)ws;                 // 2*4*4 ints        (128 B)
    float* g_N    = (float*)(ws + 128);       // 2*4*3 floats      ( 96 B)
    float* g_hist = (float*)(ws + 256);       // 2*4*297 floats    (9504 B)

    mk_init_kernel<<<10, 256, 0, stream>>>(g_N, g_hist);
    mk_bbox_kernel<<<8, 256, 0, stream>>>(seg_gen, seg_real, bbox);
    mk_hist_kernel<<<dim3(NWGX, 8), 256, 0, stream>>>(gen, realm, seg_gen, seg_real,
                                                      bbox, g_hist, g_N);
    mk_final_kernel<<<1, 64, 0, stream>>>(g_hist, g_N, out);
}